// GCNEncoder_4509715661437
// MI455X (gfx1250) — compile-verified
//
#include <hip/hip_runtime.h>
#include <hip/hip_bf16.h>
#include <cstddef>

typedef __attribute__((ext_vector_type(2))) float v2f;
typedef __attribute__((ext_vector_type(8))) float v8f;

#define CH 128

// ---------------------------------------------------------------------------
// Degree kernels: deg[i] = 1 (self loop) + #incoming edges; then dinv = rsqrt
// ---------------------------------------------------------------------------
__global__ void deg_init_kernel(float* __restrict__ deg, int n) {
    int i = blockIdx.x * blockDim.x + threadIdx.x;
    if (i < n) deg[i] = 1.0f;
}

__global__ void deg_count_kernel(const int* __restrict__ dst, float* __restrict__ deg, int e) {
    int i = blockIdx.x * blockDim.x + threadIdx.x;
    if (i < e) unsafeAtomicAdd(&deg[dst[i]], 1.0f);   // -> global_atomic_add_f32
}

__global__ void dinv_kernel(float* __restrict__ deg, int n) {
    int i = blockIdx.x * blockDim.x + threadIdx.x;
    if (i < n) deg[i] = rsqrtf(deg[i]);
}

// ---------------------------------------------------------------------------
// H = (relu?)X @ W using V_WMMA_F32_16X16X4_F32.
// One wave handles a 16-row tile; 4 waves per block => 64 rows per block.
// X tile staged in LDS (coalesced global b128 loads, ds_load_2addr_b64 A-feeds).
// Tail predication is wave-uniform: only the last of ~1563 blocks takes the
// guarded path, so the hot epilogue is straight global_store_b32s.
// ---------------------------------------------------------------------------
__global__ __launch_bounds__(128) void gemm_wmma_kernel(
    const float* __restrict__ X, const float* __restrict__ W,
    float* __restrict__ H, int nrows, int apply_relu)
{
    __shared__ float Xs[4][16 * CH];

    const int lane = threadIdx.x & 31;
    const int wave = threadIdx.x >> 5;
    const int row_base = blockIdx.x * 64 + wave * 16;
    const bool full_tile = (row_base + 16 <= nrows);   // wave-uniform

    float* xs = &Xs[wave][0];

    // Cooperative coalesced load of this wave's 16x128 tile
    if (full_tile) {
        #pragma unroll
        for (int r = 0; r < 16; ++r) {
            float4 v = *(const float4*)(X + (size_t)(row_base + r) * CH + lane * 4);
            if (apply_relu) {
                v.x = fmaxf(v.x, 0.f); v.y = fmaxf(v.y, 0.f);
                v.z = fmaxf(v.z, 0.f); v.w = fmaxf(v.w, 0.f);
            }
            *(float4*)(xs + r * CH + lane * 4) = v;
        }
    } else {
        #pragma unroll
        for (int r = 0; r < 16; ++r) {
            int row = row_base + r;
            float4 v = make_float4(0.f, 0.f, 0.f, 0.f);
            if (row < nrows)
                v = *(const float4*)(X + (size_t)row * CH + lane * 4);
            if (apply_relu) {
                v.x = fmaxf(v.x, 0.f); v.y = fmaxf(v.y, 0.f);
                v.z = fmaxf(v.z, 0.f); v.w = fmaxf(v.w, 0.f);
            }
            *(float4*)(xs + r * CH + lane * 4) = v;
        }
    }
    __syncthreads();

    // WMMA f32 16x16x4 operand layout (ISA 7.12.2):
    //   A: lane -> M = lane&15 ; VGPR0/VGPR1 = K {0,1} (lanes 0-15) / {2,3} (lanes 16-31)
    //   B/C/D: lane -> N = lane&15 ; lanes 16-31 are K+2 (B) / M+8 (C,D)
    const int m    = lane & 15;
    const int hi   = lane >> 4;     // 0 or 1
    const int colN = lane & 15;

    v8f acc[8];
    #pragma unroll
    for (int n = 0; n < 8; ++n) acc[n] = (v8f)(0.0f);

    for (int k0 = 0; k0 < CH; k0 += 4) {
        const int klo = k0 + hi * 2;
        v2f a = *(const v2f*)(xs + m * CH + klo);     // ds_load b64
        #pragma unroll
        for (int n = 0; n < 8; ++n) {
            v2f b;
            b.x = W[(size_t)klo * CH + n * 16 + colN];
            b.y = W[(size_t)(klo + 1) * CH + n * 16 + colN];
            acc[n] = __builtin_amdgcn_wmma_f32_16x16x4_f32(
                false, a, false, b, (short)0, acc[n], false, false);
        }
    }

    // D layout: VGPR r holds M=r (lanes 0-15) and M=8+r (lanes 16-31)
    if (full_tile) {
        #pragma unroll
        for (int n = 0; n < 8; ++n) {
            #pragma unroll
            for (int r = 0; r < 8; ++r) {
                int row = row_base + r + hi * 8;
                H[(size_t)row * CH + n * 16 + colN] = acc[n][r];
            }
        }
    } else if (row_base < nrows) {
        #pragma unroll
        for (int n = 0; n < 8; ++n) {
            #pragma unroll
            for (int r = 0; r < 8; ++r) {
                int row = row_base + r + hi * 8;
                if (row < nrows)
                    H[(size_t)row * CH + n * 16 + colN] = acc[n][r];
            }
        }
    }
}

// ---------------------------------------------------------------------------
// out[i][c] = h[i][c] * dinv[i]^2 + bias[c]   (self-loop + bias, full rewrite)
// ---------------------------------------------------------------------------
__global__ __launch_bounds__(256) void agg_init_kernel(
    const float* __restrict__ H, const float* __restrict__ dinv,
    const float* __restrict__ bias, float* __restrict__ out, int n)
{
    int t = blockIdx.x * blockDim.x + threadIdx.x;   // n * 32 threads
    if (t >= n * 32) return;
    int i = t >> 5;
    int c = (t & 31) * 4;
    float di = dinv[i];
    float w = di * di;
    float4 h = *(const float4*)(H + (size_t)i * CH + c);
    float4 b = *(const float4*)(bias + c);
    float4 o;
    o.x = fmaf(h.x, w, b.x);
    o.y = fmaf(h.y, w, b.y);
    o.z = fmaf(h.z, w, b.z);
    o.w = fmaf(h.w, w, b.w);
    *(float4*)(out + (size_t)i * CH + c) = o;
}

// ---------------------------------------------------------------------------
// Edge scatter: one wave32 per edge, grid-strided (~24 edges/wave).
// Lane gathers float4 of h[src], scatter-adds to out[dst] with hardware fp32
// atomics (L2-resident RMW on the 192MB L2). The next edge's source row is
// prefetched (global_prefetch_b8) to hide random-gather latency.
// ---------------------------------------------------------------------------
__global__ __launch_bounds__(256) void scatter_kernel(
    const float* __restrict__ H, const float* __restrict__ dinv,
    const int* __restrict__ src_idx, const int* __restrict__ dst_idx,
    float* __restrict__ out, int nedges)
{
    const int lane = threadIdx.x & 31;
    const int nwaves = (gridDim.x * blockDim.x) >> 5;
    int e = (blockIdx.x * blockDim.x + threadIdx.x) >> 5;

    for (; e < nedges; e += nwaves) {
        int s = src_idx[e];
        int d = dst_idx[e];

        // Prefetch next iteration's source row while we work on this one.
        int en = e + nwaves;
        if (en < nedges) {
            int sn = src_idx[en];
            __builtin_prefetch(H + (size_t)sn * CH + lane * 4, 0, 1);
        }

        float coef = dinv[s] * dinv[d];
        float4 hv = *(const float4*)(H + (size_t)s * CH + lane * 4);
        float* op = out + (size_t)d * CH + lane * 4;
        unsafeAtomicAdd(op + 0, hv.x * coef);
        unsafeAtomicAdd(op + 1, hv.y * coef);
        unsafeAtomicAdd(op + 2, hv.z * coef);
        unsafeAtomicAdd(op + 3, hv.w * coef);
    }
}

// ---------------------------------------------------------------------------
extern "C" void kernel_launch(void* const* d_in, const int* in_sizes, int n_in,
                              void* d_out, int out_size, void* d_ws, size_t ws_size,
                              hipStream_t stream) {
    const float* x    = (const float*)d_in[0];
    const int*   edge = (const int*)d_in[1];    // [2, E] (src row, dst row)
    const float* W1   = (const float*)d_in[2];
    const float* b1   = (const float*)d_in[3];
    const float* W2   = (const float*)d_in[4];
    const float* b2   = (const float*)d_in[5];
    float* out = (float*)d_out;

    const int N = in_sizes[0] / CH;
    const int E = in_sizes[1] / 2;
    const int* srcI = edge;
    const int* dstI = edge + E;

    // Workspace layout: dinv (N f32) | hbuf (N*128 f32) | tmp (N*128 f32)
    char* ws = (char*)d_ws;
    float* dinv = (float*)ws;
    size_t off  = ((size_t)N * 4 + 255) & ~(size_t)255;
    float* hbuf = (float*)(ws + off);
    float* tmp  = hbuf + (size_t)N * CH;

    const int T = 256;
    dim3 gN((N + T - 1) / T), gE((E + T - 1) / T);
    dim3 gGemm((N + 63) / 64);
    dim3 gAgg(((size_t)N * 32 + T - 1) / T);
    dim3 gScat(8192);                 // grid-stride: 65536 waves over E edges

    // Degrees (shared by both layers)
    deg_init_kernel<<<gN, T, 0, stream>>>(dinv, N);
    deg_count_kernel<<<gE, T, 0, stream>>>(dstI, dinv, E);
    dinv_kernel<<<gN, T, 0, stream>>>(dinv, N);

    // Layer 1: h = x@W1 ; tmp = h*dinv^2 + b1 ; tmp += scatter(h)
    gemm_wmma_kernel<<<gGemm, 128, 0, stream>>>(x, W1, hbuf, N, 0);
    agg_init_kernel<<<gAgg, T, 0, stream>>>(hbuf, dinv, b1, tmp, N);
    scatter_kernel<<<gScat, T, 0, stream>>>(hbuf, dinv, srcI, dstI, tmp, E);

    // Layer 2: hh = relu(tmp)@W2 ; out = hh*dinv^2 + b2 ; out += scatter(hh)
    gemm_wmma_kernel<<<gGemm, 128, 0, stream>>>(tmp, W2, hbuf, N, 1);
    agg_init_kernel<<<gAgg, T, 0, stream>>>(hbuf, dinv, b2, out, N);
    scatter_kernel<<<gScat, T, 0, stream>>>(hbuf, dinv, srcI, dstI, out, E);
}